// Head_2731599200427
// MI455X (gfx1250) — compile-verified
//
#include <hip/hip_runtime.h>
#include <hip/hip_bf16.h>

typedef __attribute__((ext_vector_type(2))) float v2f;
typedef __attribute__((ext_vector_type(8))) float v8f;
typedef __attribute__((ext_vector_type(4))) int   v4i;

#define SEQ_LEN   8192
#define TILE      16
#define CHUNK     512
#define NCHUNK    (SEQ_LEN / CHUNK)
#define TPB       256
#define WAVES_PB  8                 // 256 threads = 8 wave32
#define TWO_PI_F  6.28f             // reference uses literal 2*3.14
// scores are produced pre-multiplied by 3^-0.5 * log2(e): softmax runs in the
// log2 domain so v_exp_f32 (hardware exp2) needs no ln2 premultiply.
#define QSCALE    (0.57735026918962576f * 1.4426950408889634f)

#if __has_builtin(__builtin_amdgcn_global_load_async_to_lds_b128)
#define ASYNC_LDS 1
#else
#define ASYNC_LDS 0
#endif

__device__ __forceinline__ float fast_exp2(float x) {
#if __has_builtin(__builtin_amdgcn_exp2f)
    return __builtin_amdgcn_exp2f(x);
#else
    return exp2f(x);
#endif
}

#if ASYNC_LDS
__device__ __forceinline__ void wait_async0() {
#if __has_builtin(__builtin_amdgcn_s_wait_asynccnt)
    __builtin_amdgcn_s_wait_asynccnt(0);
#else
    asm volatile("s_wait_asynccnt 0" ::: "memory");
#endif
}
__device__ __forceinline__ void async_b128(const float4* g, float4* l) {
    __builtin_amdgcn_global_load_async_to_lds_b128(
        (__attribute__((address_space(1))) v4i*)g,
        (__attribute__((address_space(3))) v4i*)l, 0, 0);
}
#endif

// p[t] = emb[x[t]] + ( cos(6.28*t/25), sin(6.28*t/25), sin(6.28*t/5) )
__device__ __forceinline__ void embed_pos(const int* __restrict__ x,
                                          const float* __restrict__ emb,
                                          int t, float& p0, float& p1, float& p2) {
    int xi = x[t];
    float ft  = (float)t;
    float a25 = (TWO_PI_F * ft) / 25.0f;
    float a5  = (TWO_PI_F * ft) / 5.0f;
    p0 = emb[3 * xi + 0] + cosf(a25);
    p1 = emb[3 * xi + 1] + sinf(a25);
    p2 = emb[3 * xi + 2] + sinf(a5);
}

// ---- shared flash-attention pieces (inlined so LDS addrspace is inferred) ----
__device__ __forceinline__ void flash_chunk(v2f a,
                                            const float4* sK, const float4* sV,
                                            int n, int hi,
                                            float (&m)[8], float (&l)[8],
                                            float (&ax)[8], float (&ay)[8],
                                            float (&az)[8]) {
    for (int tl = 0; tl < CHUNK / TILE; ++tl) {
        const int jj = tl * TILE + n;
        float4 kr = sK[jj];
        float4 vr = sV[jj];

        v2f b;                          // B = K^T fragment (4x16), K padded w/ 0
        b.x = hi ? kr.z : kr.x;
        b.y = hi ? 0.0f : kr.y;

        v8f cz = {0.f, 0.f, 0.f, 0.f, 0.f, 0.f, 0.f, 0.f};
        v8f s = __builtin_amdgcn_wmma_f32_16x16x4_f32(
            false, a, false, b, (short)0, cz, false, false);

#pragma unroll
        for (int r = 0; r < 8; ++r) {
            float sc   = s[r];                         // log2-domain score
            float d    = sc - m[r];                    // +inf on first tile
            float e    = fast_exp2(-__builtin_fabsf(d));  // exp src mods: -|d|
            bool  nm   = d > 0.0f;                     // one VOPC mask, reused
            float corr = nm ? e : 1.0f;                // exp2(m_old - m_new)
            float p    = nm ? 1.0f : e;                // exp2(sc    - m_new)
            m[r]  = nm ? sc : m[r];                    // cndmask, no canonicalize
            l[r]  = fmaf(l[r], corr, p);
            ax[r] = fmaf(ax[r], corr, p * vr.x);
            ay[r] = fmaf(ay[r], corr, p * vr.y);
            az[r] = fmaf(az[r], corr, p * vr.z);
        }
    }
}

__device__ __forceinline__ void flash_finish(int i0, int n, int hi,
                                             float (&m)[8], float (&l)[8],
                                             float (&ax)[8], float (&ay)[8],
                                             float (&az)[8],
                                             float* __restrict__ out) {
#pragma unroll
    for (int r = 0; r < 8; ++r) {
        float mg = m[r];
#pragma unroll
        for (int msk = 1; msk < 16; msk <<= 1)
            mg = fmaxf(mg, __shfl_xor(mg, msk, 16));
        float f  = fast_exp2(m[r] - mg);
        float lg = l[r] * f, gx = ax[r] * f, gy = ay[r] * f, gz = az[r] * f;
#pragma unroll
        for (int msk = 1; msk < 16; msk <<= 1) {
            lg += __shfl_xor(lg, msk, 16);
            gx += __shfl_xor(gx, msk, 16);
            gy += __shfl_xor(gy, msk, 16);
            gz += __shfl_xor(gz, msk, 16);
        }
        if (n == 0) {
            int row = i0 + hi * 8 + r;
            float inv = 1.0f / lg;
            out[row * 3 + 0] = gx * inv;
            out[row * 3 + 1] = gy * inv;
            out[row * 3 + 2] = gz * inv;
        }
    }
}

// per-lane A fragment for rows i0..i0+15 (K padded 3->4, QSCALE folded in)
__device__ __forceinline__ v2f make_a_frag(const int* x, const float* emb,
                                           const float* Wq, int i0, int n, int hi) {
    float p0, p1, p2;
    embed_pos(x, emb, i0 + n, p0, p1, p2);
    float q0 = Wq[0] * p0 + Wq[1] * p1 + Wq[2] * p2;
    float q1 = Wq[3] * p0 + Wq[4] * p1 + Wq[5] * p2;
    float q2 = Wq[6] * p0 + Wq[7] * p1 + Wq[8] * p2;
    v2f a;
    a.x = (hi ? q2 : q0) * QSCALE;
    a.y = hi ? 0.0f : (q1 * QSCALE);
    return a;
}

// ---------------- kernel 1: materialize K/V (padded float4) into ws ----------
__global__ __launch_bounds__(TPB)
void head_prep_kernel(const int* __restrict__ x, const float* __restrict__ emb,
                      const float* __restrict__ Wk, const float* __restrict__ Wv,
                      float4* __restrict__ K4, float4* __restrict__ V4) {
    int t = blockIdx.x * TPB + threadIdx.x;
    if (t >= SEQ_LEN) return;
    float p0, p1, p2;
    embed_pos(x, emb, t, p0, p1, p2);
    K4[t] = make_float4(Wk[0] * p0 + Wk[1] * p1 + Wk[2] * p2,
                        Wk[3] * p0 + Wk[4] * p1 + Wk[5] * p2,
                        Wk[6] * p0 + Wk[7] * p1 + Wk[8] * p2, 0.0f);
    V4[t] = make_float4(Wv[0] * p0 + Wv[1] * p1 + Wv[2] * p2,
                        Wv[3] * p0 + Wv[4] * p1 + Wv[5] * p2,
                        Wv[6] * p0 + Wv[7] * p1 + Wv[8] * p2, 0.0f);
}

// ------- kernel 2: flash attention, async double-buffered K/V chunks --------
__global__ __launch_bounds__(TPB)
void head_attn_kernel(const int* __restrict__ x, const float* __restrict__ emb,
                      const float* __restrict__ Wq,
                      const float4* __restrict__ K4,
                      const float4* __restrict__ V4,
                      float* __restrict__ out) {
#if ASYNC_LDS
    __shared__ float4 sK[2][CHUNK];
    __shared__ float4 sV[2][CHUNK];
#else
    __shared__ float4 sK[1][CHUNK];
    __shared__ float4 sV[1][CHUNK];
#endif
    const int lane = threadIdx.x & 31;
    const int wave = threadIdx.x >> 5;
    const int n    = lane & 15;
    const int hi   = lane >> 4;
    const int i0   = (blockIdx.x * WAVES_PB + wave) * TILE;

    v2f a = make_a_frag(x, emb, Wq, i0, n, hi);

    float m[8], l[8], ax[8], ay[8], az[8];
#pragma unroll
    for (int r = 0; r < 8; ++r) {
        m[r] = -__builtin_inff();
        l[r] = 0.0f; ax[r] = 0.0f; ay[r] = 0.0f; az[r] = 0.0f;
    }

#if ASYNC_LDS
    // prologue: async-fill chunk 0 into buffer 0
    for (int t = threadIdx.x; t < CHUNK; t += TPB) {
        async_b128(K4 + t, &sK[0][t]);
        async_b128(V4 + t, &sV[0][t]);
    }
    wait_async0();
    __syncthreads();

    int buf = 0;
    for (int c = 0; c < NCHUNK; ++c) {
        if (c + 1 < NCHUNK) {           // prefetch next chunk into other buffer
            const float4* gk = K4 + (c + 1) * CHUNK;
            const float4* gv = V4 + (c + 1) * CHUNK;
            for (int t = threadIdx.x; t < CHUNK; t += TPB) {
                async_b128(gk + t, &sK[buf ^ 1][t]);
                async_b128(gv + t, &sV[buf ^ 1][t]);
            }
        }
        flash_chunk(a, sK[buf], sV[buf], n, hi, m, l, ax, ay, az);
        wait_async0();                  // prefetch landed (own wave)
        __syncthreads();                // all waves done reading + writing
        buf ^= 1;
    }
#else
    for (int c = 0; c < NCHUNK; ++c) {
        const int base = c * CHUNK;
        for (int t = threadIdx.x; t < CHUNK; t += TPB) {
            sK[0][t] = K4[base + t];
            sV[0][t] = V4[base + t];
        }
        __syncthreads();
        flash_chunk(a, sK[0], sV[0], n, hi, m, l, ax, ay, az);
        __syncthreads();
    }
#endif
    flash_finish(i0, n, hi, m, l, ax, ay, az, out);
}

// ---- fallback: fully fused (recomputes K/V per block), used if ws too small
__global__ __launch_bounds__(TPB)
void head_attn_fused_kernel(const int* __restrict__ x,
                            const float* __restrict__ emb,
                            const float* __restrict__ Wk,
                            const float* __restrict__ Wq,
                            const float* __restrict__ Wv,
                            float* __restrict__ out) {
    __shared__ float4 sK[CHUNK];
    __shared__ float4 sV[CHUNK];

    const int lane = threadIdx.x & 31;
    const int wave = threadIdx.x >> 5;
    const int n    = lane & 15;
    const int hi   = lane >> 4;
    const int i0   = (blockIdx.x * WAVES_PB + wave) * TILE;

    float wk[9], wv[9];
#pragma unroll
    for (int i = 0; i < 9; ++i) { wk[i] = Wk[i]; wv[i] = Wv[i]; }

    v2f a = make_a_frag(x, emb, Wq, i0, n, hi);

    float m[8], l[8], ax[8], ay[8], az[8];
#pragma unroll
    for (int r = 0; r < 8; ++r) {
        m[r] = -__builtin_inff();
        l[r] = 0.0f; ax[r] = 0.0f; ay[r] = 0.0f; az[r] = 0.0f;
    }

    for (int c = 0; c < NCHUNK; ++c) {
        const int base = c * CHUNK;
        for (int t = threadIdx.x; t < CHUNK; t += TPB) {
            float e0, e1, e2;
            embed_pos(x, emb, base + t, e0, e1, e2);
            sK[t] = make_float4(wk[0] * e0 + wk[1] * e1 + wk[2] * e2,
                                wk[3] * e0 + wk[4] * e1 + wk[5] * e2,
                                wk[6] * e0 + wk[7] * e1 + wk[8] * e2, 0.0f);
            sV[t] = make_float4(wv[0] * e0 + wv[1] * e1 + wv[2] * e2,
                                wv[3] * e0 + wv[4] * e1 + wv[5] * e2,
                                wv[6] * e0 + wv[7] * e1 + wv[8] * e2, 0.0f);
        }
        __syncthreads();
        flash_chunk(a, sK, sV, n, hi, m, l, ax, ay, az);
        __syncthreads();
    }
    flash_finish(i0, n, hi, m, l, ax, ay, az, out);
}

extern "C" void kernel_launch(void* const* d_in, const int* in_sizes, int n_in,
                              void* d_out, int out_size, void* d_ws, size_t ws_size,
                              hipStream_t stream) {
    (void)in_sizes; (void)n_in; (void)out_size;
    const int*   x   = (const int*)  d_in[0];
    const float* emb = (const float*)d_in[1];
    const float* Wk  = (const float*)d_in[2];
    const float* Wq  = (const float*)d_in[3];
    const float* Wv  = (const float*)d_in[4];
    float* out = (float*)d_out;

    const int blocks = (SEQ_LEN / TILE) / WAVES_PB;   // 64 blocks x 8 waves
    const size_t need = (size_t)2 * SEQ_LEN * sizeof(float4);   // 256 KB

    if (ws_size >= need) {
        float4* K4 = (float4*)d_ws;
        float4* V4 = K4 + SEQ_LEN;
        head_prep_kernel<<<SEQ_LEN / TPB, TPB, 0, stream>>>(x, emb, Wk, Wv, K4, V4);
        head_attn_kernel<<<blocks, TPB, 0, stream>>>(x, emb, Wq, K4, V4, out);
    } else {
        head_attn_fused_kernel<<<blocks, TPB, 0, stream>>>(x, emb, Wk, Wq, Wv, out);
    }
}